// SSIM3DApproximation_56968446214326
// MI455X (gfx1250) — compile-verified
//
#include <hip/hip_runtime.h>
#include <hip/hip_bf16.h>
#include <stdint.h>

#define H_IMG 512
#define W_IMG 512
#define D_IMG 64
#define TILE_H 16
#define TILE_W 64
#define RH (TILE_H + 10)   // 26 region rows
#define RW (TILE_W + 10)   // 74 region cols
#define RWP 76             // padded LDS row stride (floats)
#define NTHREADS 256
#define C1_CONST 1.0e-4f
#define C2_CONST 9.0e-4f

// ---------------------------------------------------------------------------
// Kernel 1: one block = one 16x64 output tile of one slice.
// Async-copies the (26 x 74) halo region of x and y into LDS
// (global_load_async_to_lds_b32, ASYNCcnt), does separable 11-tap box sums
// for the 5 SSIM moments, evaluates the SSIM map, and writes one partial sum
// per block.
// ---------------------------------------------------------------------------
__global__ __launch_bounds__(NTHREADS)
void ssim_tile_kernel(const float* __restrict__ x,
                      const float* __restrict__ y,
                      const float* __restrict__ window,
                      float* __restrict__ partials) {
    __shared__ float xs[RH * RWP];
    __shared__ float ys[RH * RWP];
    __shared__ float csx [TILE_H * RWP];
    __shared__ float csy [TILE_H * RWP];
    __shared__ float csxx[TILE_H * RWP];
    __shared__ float csyy[TILE_H * RWP];
    __shared__ float csxy[TILE_H * RWP];
    __shared__ float red[NTHREADS];

    const int tid   = threadIdx.x;
    const int tw    = blockIdx.x;        // 0..7   column tile
    const int th    = blockIdx.y;        // 0..31  row tile
    const int slice = blockIdx.z;        // 0..63

    const float w0 = window[0];          // uniform window weight (1/121)

    const int base_h = th * TILE_H - 5;
    const int base_w = tw * TILE_W - 5;

    const float* __restrict__ xsl = x + (size_t)slice * (H_IMG * W_IMG);
    const float* __restrict__ ysl = y + (size_t)slice * (H_IMG * W_IMG);

    // ---- Phase 1: async global -> LDS tile load (zero-pad out of bounds) ----
    for (int t = tid; t < RH * RW; t += NTHREADS) {
        const int rr = t / RW;
        const int cc = t - rr * RW;
        const int h  = base_h + rr;
        const int w  = base_w + cc;
        const int li = rr * RWP + cc;
        if (h >= 0 && h < H_IMG && w >= 0 && w < W_IMG) {
            const float* gx = xsl + (size_t)h * W_IMG + w;
            const float* gy = ysl + (size_t)h * W_IMG + w;
            unsigned lox = (unsigned)(uintptr_t)(&xs[li]);
            unsigned loy = (unsigned)(uintptr_t)(&ys[li]);
            asm volatile("global_load_async_to_lds_b32 %0, %1, off"
                         :: "v"(lox), "v"(gx) : "memory");
            asm volatile("global_load_async_to_lds_b32 %0, %1, off"
                         :: "v"(loy), "v"(gy) : "memory");
        } else {
            xs[li] = 0.0f;
            ys[li] = 0.0f;
        }
    }
    asm volatile("s_wait_asynccnt 0" ::: "memory");
    __syncthreads();

    // ---- Phase 2: 11-tap column sums for 5 moments (LDS -> LDS) ----
    for (int t = tid; t < TILE_H * RW; t += NTHREADS) {
        const int r = t / RW;
        const int c = t - r * RW;
        float sx = 0.f, sy = 0.f, sxx = 0.f, syy = 0.f, sxy = 0.f;
#pragma unroll
        for (int i = 0; i < 11; ++i) {
            const float xv = xs[(r + i) * RWP + c];
            const float yv = ys[(r + i) * RWP + c];
            sx += xv;
            sy += yv;
            sxx = fmaf(xv, xv, sxx);
            syy = fmaf(yv, yv, syy);
            sxy = fmaf(xv, yv, sxy);
        }
        const int o = r * RWP + c;
        csx[o]  = sx;
        csy[o]  = sy;
        csxx[o] = sxx;
        csyy[o] = syy;
        csxy[o] = sxy;
    }
    __syncthreads();

    // ---- Phase 3: 11-tap row sums + SSIM map + local accumulation ----
    float acc = 0.0f;
    for (int t = tid; t < TILE_H * TILE_W; t += NTHREADS) {
        const int r = t >> 6;           // /64
        const int w = t & 63;           // %64
        float Sx = 0.f, Sy = 0.f, Sxx = 0.f, Syy = 0.f, Sxy = 0.f;
#pragma unroll
        for (int j = 0; j < 11; ++j) {
            const int o = r * RWP + w + j;
            Sx  += csx[o];
            Sy  += csy[o];
            Sxx += csxx[o];
            Syy += csyy[o];
            Sxy += csxy[o];
        }
        const float mux  = Sx  * w0;
        const float muy  = Sy  * w0;
        const float ex2  = Sxx * w0;
        const float ey2  = Syy * w0;
        const float exy  = Sxy * w0;
        const float mux2 = mux * mux;
        const float muy2 = muy * muy;
        const float muxy = mux * muy;
        const float sx2  = ex2 - mux2;
        const float sy2  = ey2 - muy2;
        const float sxy  = exy - muxy;
        const float num  = (2.0f * muxy + C1_CONST) * (2.0f * sxy + C2_CONST);
        const float den  = (mux2 + muy2 + C1_CONST) * (sx2 + sy2 + C2_CONST);
        acc += num / den;
    }

    // ---- Block reduction ----
    red[tid] = acc;
    __syncthreads();
#pragma unroll
    for (int s = NTHREADS / 2; s > 0; s >>= 1) {
        if (tid < s) red[tid] += red[tid + s];
        __syncthreads();
    }
    if (tid == 0) {
        const int blin = (int)blockIdx.z * (gridDim.x * gridDim.y)
                       + (int)blockIdx.y * gridDim.x + (int)blockIdx.x;
        partials[blin] = red[0];
    }
}

// ---------------------------------------------------------------------------
// Kernel 2: reduce per-block partials -> final scalar 1 - mean(ssim_map)
// ---------------------------------------------------------------------------
__global__ __launch_bounds__(NTHREADS)
void ssim_final_reduce(const float* __restrict__ partials,
                       float* __restrict__ out,
                       int n, float inv_n_pixels) {
    __shared__ float red[NTHREADS];
    float a = 0.0f;
    for (int i = threadIdx.x; i < n; i += NTHREADS) a += partials[i];
    red[threadIdx.x] = a;
    __syncthreads();
#pragma unroll
    for (int s = NTHREADS / 2; s > 0; s >>= 1) {
        if (threadIdx.x < s) red[threadIdx.x] += red[threadIdx.x + s];
        __syncthreads();
    }
    if (threadIdx.x == 0) out[0] = 1.0f - red[0] * inv_n_pixels;
}

extern "C" void kernel_launch(void* const* d_in, const int* in_sizes, int n_in,
                              void* d_out, int out_size, void* d_ws, size_t ws_size,
                              hipStream_t stream) {
    (void)in_sizes; (void)n_in; (void)out_size; (void)ws_size;
    const float* x      = (const float*)d_in[0];
    const float* y      = (const float*)d_in[1];
    const float* window = (const float*)d_in[2];
    float* out          = (float*)d_out;
    float* partials     = (float*)d_ws;   // 8*32*64 = 16384 floats = 64 KB

    dim3 grid(W_IMG / TILE_W, H_IMG / TILE_H, D_IMG);   // (8, 32, 64)
    ssim_tile_kernel<<<grid, NTHREADS, 0, stream>>>(x, y, window, partials);

    const int nPartials = (W_IMG / TILE_W) * (H_IMG / TILE_H) * D_IMG;
    const float invN = 1.0f / (float)((size_t)D_IMG * H_IMG * W_IMG);
    ssim_final_reduce<<<1, NTHREADS, 0, stream>>>(partials, out, nPartials, invN);
}